// BoundaryLoss_5789615915056
// MI455X (gfx1250) — compile-verified
//
#include <hip/hip_runtime.h>
#include <math.h>

#define CLASSES 19
#define HGT 512
#define WID 512
#define BATCH 8
#define HW (HGT * WID)
#define TILE 32
#define HALO 19
#define TDIM (TILE + 2 * HALO)   // 70
#define TPITCH 72                // padded pitch
#define FULLMASK ((1u << CLASSES) - 1u)

typedef __attribute__((ext_vector_type(2))) float v2f;
typedef __attribute__((ext_vector_type(8))) float v8f;

// ---------------------------------------------------------------------------
// Kernel 1: fused softmax + bit-packed Chebyshev ring scan + block reduction.
// One block = one 32x32 tile of one batch image. 256 threads, 4 pixels each.
// ---------------------------------------------------------------------------
__global__ __launch_bounds__(256) void boundary_main(
    const float* __restrict__ pred, const int* __restrict__ target,
    float* __restrict__ partial) {
  __shared__ unsigned tile[TDIM * TPITCH];  // 70*72*4 = 20160 B
  __shared__ float red[256];

  const int tx = blockIdx.x, ty = blockIdx.y, b = blockIdx.z;
  const int x0 = tx * TILE - HALO, y0 = ty * TILE - HALO;
  const int t = threadIdx.x;

  // Stage halo tile of one-hot class words (OOB -> 0, matching SAME padding).
  const int* tgt_b = target + (size_t)b * HW;
  for (int i = t; i < TDIM * TDIM; i += 256) {
    int ly = i / TDIM, lx = i - ly * TDIM;
    int gy = y0 + ly, gx = x0 + lx;
    unsigned w = 0u;
    if ((unsigned)gy < (unsigned)HGT && (unsigned)gx < (unsigned)WID)
      w = 1u << (tgt_b[(size_t)gy * WID + gx] & 31);
    tile[ly * TPITCH + lx] = w;
  }
  __syncthreads();

  const int lx = t & 31;        // lane -> x (coalesced pred reads)
  const int lyb = t >> 5;       // 0..7
  float accT = 0.f;

  for (int k = 0; k < 4; ++k) {
    const int ly = lyb + 8 * k;                  // 0..31
    const int gx = tx * TILE + lx, gy = ty * TILE + ly;
    const float* pp = pred + (size_t)b * CLASSES * HW + (size_t)gy * WID + gx;

    // Softmax denominator (pass over C=19 with stride HW).
    float m = pp[0];
#pragma unroll
    for (int c = 1; c < CLASSES; ++c) m = fmaxf(m, pp[(size_t)c * HW]);
    float s = 0.f;
#pragma unroll
    for (int c = 0; c < CLASSES; ++c) s += __expf(pp[(size_t)c * HW] - m);
    const float invs = 1.f / s;

    const int cx = lx + HALO, cy = ly + HALO;
    const unsigned w0 = tile[cy * TPITCH + cx];
    const int c0 = __builtin_ctz(w0);
    unsigned found = w0;
    float S = __expf(pp[(size_t)c0 * HW] - m) * invs;  // prob of own class
    float acc = 1.f - S;                               // r = 0 term

    for (int r = 1; r <= 19; ++r) {
      // OR of the Chebyshev ring at radius r (8r cells).
      unsigned ro = 0u;
      const unsigned* top = &tile[(cy - r) * TPITCH + cx];
      const unsigned* bot = &tile[(cy + r) * TPITCH + cx];
      for (int dx = -r; dx <= r; ++dx) ro |= top[dx] | bot[dx];
      for (int dy = -r + 1; dy <= r - 1; ++dy) {
        const unsigned* row = &tile[(cy + dy) * TPITCH + cx];
        ro |= row[-r] | row[r];
      }
      // Isolated-pixel correction: vanishes at t=0, reactivates at t=1 -> +1.
      if (r == 1 && !(ro & w0)) acc += __expf(pp[(size_t)c0 * HW] - m) * invs;

      unsigned newly = ro & ~found;
      while (newly) {
        int c = __builtin_ctz(newly);
        newly &= newly - 1u;
        S += __expf(pp[(size_t)c * HW] - m) * invs;  // L0-hot reload
      }
      found |= ro;
      acc += 1.f - S;  // term r of min(D,20) = sum_r [D > r]
      if (found == FULLMASK) {         // early exit: remaining terms identical
        acc += (float)(19 - r) * (1.f - S);
        break;
      }
    }
    accT += acc;
  }

  // Deterministic block reduction.
  red[t] = accT;
  __syncthreads();
  for (int off = 128; off > 0; off >>= 1) {
    if (t < off) red[t] += red[t + off];
    __syncthreads();
  }
  if (t == 0) partial[((size_t)b * gridDim.y + ty) * gridDim.x + tx] = red[0];
}

// ---------------------------------------------------------------------------
// Kernel 2: single-wave32 WMMA reduction of the 2048 block partials.
// B = all-ones => D[m,n] = sum_k A[m,k] + C[m,n]; sum of all 256 D slots is
// 16x the grand total, independent of operand packing. Exact f32, determinstic.
// ---------------------------------------------------------------------------
__global__ __launch_bounds__(32) void reduce_final(
    const float* __restrict__ partial, float* __restrict__ out) {
  const int lane = threadIdx.x;  // 0..31, EXEC all ones (WMMA requirement)
  float total;
#if __has_builtin(__builtin_amdgcn_wmma_f32_16x16x4_f32)
  v2f ones; ones.x = 1.f; ones.y = 1.f;
  v8f c = {};
  for (int chunk = 0; chunk < 2048; chunk += 64) {
    const float* p = partial + chunk + lane * 2;
    v2f a; a.x = p[0]; a.y = p[1];
    c = __builtin_amdgcn_wmma_f32_16x16x4_f32(
        /*neg_a=*/false, a, /*neg_b=*/false, ones,
        /*c_mod=*/(short)0, c, /*reuse_a=*/false, /*reuse_b=*/false);
  }
  float s8 = c[0] + c[1] + c[2] + c[3] + c[4] + c[5] + c[6] + c[7];
#pragma unroll
  for (int off = 16; off > 0; off >>= 1) s8 += __shfl_xor(s8, off, 32);
  total = s8 * (1.f / 16.f);  // 16 replicated columns
#else
  float s = 0.f;
  for (int i = lane; i < 2048; i += 32) s += partial[i];
#pragma unroll
  for (int off = 16; off > 0; off >>= 1) s += __shfl_xor(s, off, 32);
  total = s;
#endif
  if (lane == 0)
    out[0] = total * (1.0f / ((float)BATCH * CLASSES * HGT * WID));  // WEIGHT=1
}

extern "C" void kernel_launch(void* const* d_in, const int* in_sizes, int n_in,
                              void* d_out, int out_size, void* d_ws,
                              size_t ws_size, hipStream_t stream) {
  const float* pred = (const float*)d_in[0];
  const int* target = (const int*)d_in[1];
  float* out = (float*)d_out;
  float* partial = (float*)d_ws;  // 16x16x8 = 2048 floats

  dim3 grid(WID / TILE, HGT / TILE, BATCH);  // (16,16,8)
  boundary_main<<<grid, 256, 0, stream>>>(pred, target, partial);
  reduce_final<<<1, 32, 0, stream>>>(partial, out);
}